// Embedding_61022895342169
// MI455X (gfx1250) — compile-verified
//
#include <hip/hip_runtime.h>
#include <hip/hip_bf16.h>

// Embedding row-gather: out[r, :] = weight[idx[r], :]
//   idx    : int32  [N_IDX]           (d_in[0])
//   weight : float  [N_EMB, 128]      (d_in[1])
//   out    : float  [N_IDX, 128]      (d_out)
//
// Memory-bound (0 FLOPs): roofline = 512 MB / 23.3 TB/s ~= 22 us.
// One wave32 owns 32 consecutive rows:
//   - 1 coalesced b32 load grabs all 32 row indices (one per lane)
//   - v_readlane_b32 (imm lane) broadcasts each index to an SGPR, so row
//     source/dest addresses are scalar (saddr-form global b128 ops)
//   - unroll-8 software pipeline: 8 global_load_b128 in flight per wave,
//     then 8 non-temporal global_store_b128
// Table reads stay regular-temporal (~21% repeat rows hit the 192 MB L2);
// the 256 MB write-once output stream is NT so it never evicts the table.

typedef float v4f __attribute__((ext_vector_type(4)));

#define D_EMB         128   // floats per row; 128 / 32 lanes = 1 float4/lane
#define ROWS_PER_WAVE 32
#define UNROLL        8

__global__ __launch_bounds__(256) void
embedding_gather_kernel(const int* __restrict__ idx,
                        const float* __restrict__ weight,
                        float* __restrict__ out,
                        int n_rows) {
    const int lane          = threadIdx.x & 31;
    const int wavesPerBlock = blockDim.x >> 5;
    const int wave          = blockIdx.x * wavesPerBlock + (threadIdx.x >> 5);
    const int base          = wave * ROWS_PER_WAVE;
    if (base >= n_rows) return;

    // One coalesced transaction: 32 row indices for this wave (one per lane).
    const int r_lane = base + lane;
    const int myIdx  = (r_lane < n_rows) ? idx[r_lane] : 0;

    if (base + ROWS_PER_WAVE <= n_rows) {
        // Hot path: full tile, no guards.
        #pragma unroll
        for (int j = 0; j < ROWS_PER_WAVE; j += UNROLL) {
            v4f v[UNROLL];
            // Issue UNROLL independent b128 gathers (8 loads in flight).
            #pragma unroll
            for (int k = 0; k < UNROLL; ++k) {
                const int row = __builtin_amdgcn_readlane(myIdx, j + k);
                const v4f* __restrict__ src =
                    (const v4f*)(weight + (size_t)row * D_EMB) + lane;
                v[k] = *src;
            }
            // Drain with non-temporal streaming stores.
            #pragma unroll
            for (int k = 0; k < UNROLL; ++k) {
                const int r = base + j + k;
                v4f* __restrict__ dst = (v4f*)(out + (size_t)r * D_EMB) + lane;
                __builtin_nontemporal_store(v[k], dst);
            }
        }
    } else {
        // Tail tile: per-row wave-uniform guards (scalar branches).
        #pragma unroll
        for (int j = 0; j < ROWS_PER_WAVE; ++j) {
            const int r = base + j;
            if (r < n_rows) {
                const int row = __builtin_amdgcn_readlane(myIdx, j);
                const v4f* __restrict__ src =
                    (const v4f*)(weight + (size_t)row * D_EMB) + lane;
                v4f v = *src;
                v4f* __restrict__ dst = (v4f*)(out + (size_t)r * D_EMB) + lane;
                __builtin_nontemporal_store(v, dst);
            }
        }
    }
}

extern "C" void kernel_launch(void* const* d_in, const int* in_sizes, int n_in,
                              void* d_out, int out_size, void* d_ws, size_t ws_size,
                              hipStream_t stream) {
    const int*   idx    = (const int*)d_in[0];
    const float* weight = (const float*)d_in[1];
    float*       out    = (float*)d_out;

    const int n_rows = in_sizes[0];  // N_IDX = 500,000

    const int threads       = 256;                 // 8 wave32s per block
    const int wavesPerBlock = threads / 32;
    const int nWaves        = (n_rows + ROWS_PER_WAVE - 1) / ROWS_PER_WAVE;
    int blocks = (nWaves + wavesPerBlock - 1) / wavesPerBlock;
    if (blocks < 1) blocks = 1;

    embedding_gather_kernel<<<blocks, threads, 0, stream>>>(idx, weight, out, n_rows);
}